// InteractionBlock_58557584114117
// MI455X (gfx1250) — compile-verified
//
#include <hip/hip_runtime.h>
#include <math.h>

// ---------------------------------------------------------------------------
// SchNet-style interaction block for gfx1250 (MI455X), full fp32 via
// V_WMMA_F32_16X16X4_F32.  All dims are multiples of 16 so every WMMA tile is
// full and EXEC is all-ones around matrix ops (ISA 7.12 requirement).
//
// Round-2 changes:
//  * edge kernel processes 64 edges/block (4 M-tiles per wave) so each
//    fw1/fw2 B-fragment load is reused by 4 WMMAs -> 4x less L2 weight traffic
//  * LDS tiles padded (+4 floats/row) to kill 16-way bank conflicts on the
//    A-fragment ds_loads (row strides 256 and 64 are 0 mod 64 banks)
// ---------------------------------------------------------------------------

typedef __attribute__((ext_vector_type(2))) float v2f;
typedef __attribute__((ext_vector_type(8))) float v8f;

#define HIDDEN 256
#define NGAUSS 64
#define NFILT  256

#define TE     64              // edges per block in the edge kernel
#define PADB   (NGAUSS + 4)    // padded row stride for basis tile (banks: 4*l)
#define PADA   (NFILT + 4)     // padded row stride for activation tile

__device__ __forceinline__ float sspf(float x) {
  // softplus(x) - log(2), numerically stable
  float ax = fabsf(x);
  return fmaxf(x, 0.0f) + log1pf(expf(-ax)) - 0.69314718055994530942f;
}

__device__ __forceinline__ void atomic_add_f32(float* p, float v) {
  // non-returning relaxed agent-scope add -> GLOBAL_ATOMIC_ADD_F32
  __hip_atomic_fetch_add(p, v, __ATOMIC_RELAXED, __HIP_MEMORY_SCOPE_AGENT);
}

__global__ void zero_f32(float* __restrict__ p, int n) {
  int i = blockIdx.x * blockDim.x + threadIdx.x;
  if (i < n) p[i] = 0.0f;
}

// ---------------------------------------------------------------------------
// D[m, n] = act( sum_k A[m,k] * Wt[n,k] + bias[n] )
// A : [M, K] row-major,  Wt : [NC, K] row-major (i.e. applies Wt^T),
// D : [M, 256] row-major.
// grid.x = M/16, grid.y = NC/64, block = 128 threads (4 waves, 16 cols each).
// ---------------------------------------------------------------------------
template <int ACT>
__global__ void gemm_wmma_f32(const float* __restrict__ A,
                              const float* __restrict__ Wt,
                              const float* __restrict__ bias,
                              float* __restrict__ D,
                              int K) {
  const int lane = threadIdx.x & 31;
  const int wave = threadIdx.x >> 5;
  const int m0 = blockIdx.x * 16;
  const int n0 = blockIdx.y * 64 + wave * 16;
  const int l  = lane & 15;             // M (for A) / N (for B) within tile
  const int kk = (lane >> 4) << 1;      // K sub-pair: lanes 16..31 take K+2,K+3

  const float* Arow = A  + (size_t)(m0 + l) * K;
  const float* Wrow = Wt + (size_t)(n0 + l) * K;

  v8f acc = {};
  for (int k0 = 0; k0 < K; k0 += 4) {
    v2f a, b;
    a.x = Arow[k0 + kk];
    a.y = Arow[k0 + kk + 1];
    b.x = Wrow[k0 + kk];
    b.y = Wrow[k0 + kk + 1];
    acc = __builtin_amdgcn_wmma_f32_16x16x4_f32(false, a, false, b,
                                                (short)0, acc, false, false);
  }

  const int col   = n0 + l;
  const int rbase = (lane >> 4) << 3;   // lanes 16..31 hold rows M+8..M+15
  const float bv  = bias ? bias[col] : 0.0f;
#pragma unroll
  for (int r = 0; r < 8; ++r) {
    float v = acc[r] + bv;
    if (ACT) v = sspf(v);
    D[(size_t)(m0 + rbase + r) * 256 + col] = v;
  }
}

// ---------------------------------------------------------------------------
// Fused edge pipeline, one block per 64 edges, 512 threads (16 waves).
// Each wave owns 16 output columns and iterates 4 edge M-tiles, so every
// B-fragment (weight) load feeds 4 WMMAs.
//   t  = ssp(basis @ fw1^T + fb1)            (WMMA GEMM1, K=64,  -> LDS)
//   W  = (t @ fw2^T + fb2) * cutoff(e)       (WMMA GEMM2, K=256)
//   atomically: agg[dst] += h[src] * W
// ---------------------------------------------------------------------------
__global__ void __launch_bounds__(512)
edge_filter_scatter(const float* __restrict__ basis,   // [E,64]
                    const float* __restrict__ e_ji,    // [E]
                    const long long* __restrict__ ji,  // [2,E]
                    const float* __restrict__ fw1,
                    const float* __restrict__ fb1,
                    const float* __restrict__ fw2,
                    const float* __restrict__ fb2,
                    const float* __restrict__ h,       // [N,256]
                    float* __restrict__ agg,           // [N,256]
                    int E) {
  __shared__ float sBasis[TE * PADB];   // 64 x 68  -> 17.0 KB
  __shared__ float sAct[TE * PADA];     // 64 x 260 -> 65.0 KB
  __shared__ float sC[TE];
  __shared__ int   sSrc[TE];
  __shared__ int   sDst[TE];

  const int tid  = threadIdx.x;
  const int lane = tid & 31;
  const int wave = tid >> 5;
  const int e0   = blockIdx.x * TE;

  // stage basis tile + per-edge scalars into LDS
  for (int i = tid; i < TE * NGAUSS; i += 512) {
    int r = i >> 6, g = i & 63;
    sBasis[r * PADB + g] = basis[(size_t)(e0 + r) * NGAUSS + g];
  }
  if (tid < TE) {
    int e = e0 + tid;
    sC[tid]   = 0.25f * (cosf(e_ji[e] * 3.14159265358979323846f) + 1.0f);
    sSrc[tid] = (int)ji[e];
    sDst[tid] = (int)ji[(size_t)E + e];
  }
  __syncthreads();

  const int l     = lane & 15;
  const int kk    = (lane >> 4) << 1;
  const int n0    = wave * 16;
  const int col   = n0 + l;
  const int rbase = (lane >> 4) << 3;

  // ---- GEMM1: 4 x ([16 x 64] @ fw1^T) -> ssp -> sAct cols [n0, n0+16) ----
  {
    v8f acc[4] = {};
    const float* Wrow = fw1 + (size_t)col * NGAUSS;
    for (int k0 = 0; k0 < NGAUSS; k0 += 4) {
      v2f b;
      b.x = Wrow[k0 + kk];
      b.y = Wrow[k0 + kk + 1];
#pragma unroll
      for (int t = 0; t < 4; ++t) {
        v2f a;
        const int arow = (t * 16 + l) * PADB + k0 + kk;
        a.x = sBasis[arow];
        a.y = sBasis[arow + 1];
        acc[t] = __builtin_amdgcn_wmma_f32_16x16x4_f32(false, a, false, b,
                                                       (short)0, acc[t],
                                                       false, false);
      }
    }
    const float bv = fb1[col];
#pragma unroll
    for (int t = 0; t < 4; ++t) {
#pragma unroll
      for (int r = 0; r < 8; ++r) {
        sAct[(t * 16 + rbase + r) * PADA + col] = sspf(acc[t][r] + bv);
      }
    }
  }
  __syncthreads();

  // ---- GEMM2: 4 x ([16 x 256] @ fw2^T), scale by cutoff, gather h, scatter ----
  {
    v8f acc[4] = {};
    const float* Wrow = fw2 + (size_t)col * NFILT;
    for (int k0 = 0; k0 < NFILT; k0 += 4) {
      v2f b;
      b.x = Wrow[k0 + kk];
      b.y = Wrow[k0 + kk + 1];
#pragma unroll
      for (int t = 0; t < 4; ++t) {
        v2f a;
        const int arow = (t * 16 + l) * PADA + k0 + kk;
        a.x = sAct[arow];
        a.y = sAct[arow + 1];
        acc[t] = __builtin_amdgcn_wmma_f32_16x16x4_f32(false, a, false, b,
                                                       (short)0, acc[t],
                                                       false, false);
      }
    }
    const float bv = fb2[col];
#pragma unroll
    for (int t = 0; t < 4; ++t) {
#pragma unroll
      for (int r = 0; r < 8; ++r) {
        const int row   = t * 16 + rbase + r;
        const float w   = (acc[t][r] + bv) * sC[row];
        const float msg = h[(size_t)sSrc[row] * HIDDEN + col] * w;
        atomic_add_f32(&agg[(size_t)sDst[row] * NFILT + col], msg);
      }
    }
  }
}

// ---------------------------------------------------------------------------
extern "C" void kernel_launch(void* const* d_in, const int* in_sizes, int n_in,
                              void* d_out, int out_size, void* d_ws, size_t ws_size,
                              hipStream_t stream) {
  const float*     x     = (const float*)d_in[0];
  const long long* ji    = (const long long*)d_in[1];
  const float*     e_ji  = (const float*)d_in[2];
  const float*     basis = (const float*)d_in[3];
  const float*     fw1   = (const float*)d_in[4];
  const float*     fb1   = (const float*)d_in[5];
  const float*     fw2   = (const float*)d_in[6];
  const float*     fb2   = (const float*)d_in[7];
  const float*     w1    = (const float*)d_in[8];
  const float*     w2    = (const float*)d_in[9];
  const float*     b2    = (const float*)d_in[10];
  const float*     w3    = (const float*)d_in[11];
  const float*     b3    = (const float*)d_in[12];

  const int N = in_sizes[0] / HIDDEN;   // 10000 (multiple of 16)
  const int E = in_sizes[2];            // 320000 (multiple of 64)

  float* h   = (float*)d_ws;                    // [N,256], later reused for t
  float* agg = h + (size_t)N * NFILT;           // [N,256]
  float* out = (float*)d_out;

  const int nAgg = N * NFILT;
  zero_f32<<<(nAgg + 255) / 256, 256, 0, stream>>>(agg, nAgg);

  // h = x @ w1^T   (no bias, no activation)
  gemm_wmma_f32<0><<<dim3(N / 16, NFILT / 64), 128, 0, stream>>>(x, w1, nullptr, h, HIDDEN);

  // edge filter network + message + segment-sum (atomics)
  edge_filter_scatter<<<E / TE, 512, 0, stream>>>(basis, e_ji, ji,
                                                  fw1, fb1, fw2, fb2,
                                                  h, agg, E);

  // t = ssp(agg @ w2^T + b2)   (reuse h buffer as t)
  gemm_wmma_f32<1><<<dim3(N / 16, HIDDEN / 64), 128, 0, stream>>>(agg, w2, b2, h, NFILT);

  // out = t @ w3^T + b3
  gemm_wmma_f32<0><<<dim3(N / 16, HIDDEN / 64), 128, 0, stream>>>(h, w3, b3, out, HIDDEN);
}